// LocallyLowRank_13537736917131
// MI455X (gfx1250) — compile-verified
//
#include <hip/hip_runtime.h>
#include <hip/hip_bf16.h>
#include <stdint.h>

// Locally-low-rank patch denoise for MI455X (gfx1250, wave32).
// - Pair two 8x256 patch matrices into one 16x256 "M" per wave.
// - Stage M into LDS with GLOBAL_LOAD_ASYNC_TO_LDS_B32 (ASYNCcnt path).
// - Gram G16 = M*M^T and reconstruct W16*M via V_WMMA_F32_16X16X4_F32.
// - 8x8 Jacobi eigensolve (1 lane/patch) for the SVD soft-threshold.
// - Overlap-add with global_atomic_add_f32, then shifted weight divide.

#define NN      4
#define AA      8
#define HH      512
#define WW      512
#define NB      63              // (512-16)/8 + 1
#define PATCHES (NN * NB * NB)  // 15876
#define PAIRS   (PATCHES / 2)   // 7938
#define WPB     2               // waves per block (LDS budget: 36 KB)
#define THRESH  2.0f
// round(np.random.default_rng(1000).uniform(-3,3)) -- fixed host-side constant
// (compile-only environment; value computed offline).
#define SHIFT   1

typedef __attribute__((ext_vector_type(2))) float v2f;
typedef __attribute__((ext_vector_type(8))) float v8f;

__device__ __forceinline__ int wrap512(int v) { return v & 511; }

// Async copy of one f32 from global to a per-lane LDS destination.
// dsaddr (VDST) is the wave-relative LDS byte offset == low 32 bits of the
// generic pointer to the __shared__ element.
__device__ __forceinline__ void async_g2lds_b32(const float* gaddr, void* lds) {
    unsigned lds_off = (unsigned)(uintptr_t)lds;
    asm volatile("global_load_async_to_lds_b32 %0, %1, off"
                 :
                 : "v"(lds_off), "v"(gaddr)
                 : "memory");
}

__device__ __forceinline__ void wait_asynccnt0() {
    asm volatile("s_wait_asynccnt 0x0" ::: "memory");
}

__global__ void llr_zero_kernel(float* __restrict__ out, int n) {
    int i = blockIdx.x * blockDim.x + threadIdx.x;
    if (i < n) out[i] = 0.0f;
}

__global__ __launch_bounds__(WPB * 32)
void llr_main_kernel(const float* __restrict__ x, float* __restrict__ out) {
    __shared__ float sM[WPB][16][256];  // paired patch matrix M (row = a + 8*patch)
    __shared__ float sG[WPB][16][16];   // Gram G16 = M M^T
    __shared__ float sW[WPB][16][16];   // block-diagonal shrinkage operator W16

    const int wid  = threadIdx.x >> 5;
    const int lane = threadIdx.x & 31;
    const int pair = blockIdx.x * WPB + wid;

    // Decode the two patches of this wave.
    const int q0 = 2 * pair, q1 = q0 + 1;
    const int n0 = q0 / (NB * NB), r0 = q0 % (NB * NB);
    const int n1 = q1 / (NB * NB), r1 = q1 % (NB * NB);
    const int h00 = (r0 / NB) * 8, w00 = (r0 % NB) * 8;
    const int h01 = (r1 / NB) * 8, w01 = (r1 % NB) * 8;

    // ---- Stage M (16 x 256) into LDS via async-to-LDS (no VGPR round-trip).
    // Rolled coordinate p reads x[(p - SHIFT) mod 512]; per-element wrap forces
    // scattered per-lane addresses, which the async path supports directly.
    #pragma unroll 4
    for (int it = 0; it < 64; ++it) {
        int e   = it * 64 + lane * 2;          // two consecutive cols per lane
        int row = e >> 8;
        int col = e & 255;
        int pt  = row >> 3;
        int a   = row & 7;
        int n   = pt ? n1 : n0;
        int hb  = pt ? h01 : h00;
        int wb  = pt ? w01 : w00;
        int dy  = col >> 4, dx = col & 15;     // dx even, dx+1 <= 15
        int hp  = wrap512(hb + dy - SHIFT);
        int wp0 = wrap512(wb + dx - SHIFT);
        int wp1 = wrap512(wb + dx + 1 - SHIFT);
        const float* base = x + (((size_t)(n * AA + a) * HH + hp) * WW);
        async_g2lds_b32(base + wp0, &sM[wid][row][col]);
        async_g2lds_b32(base + wp1, &sM[wid][row][col + 1]);
    }
    wait_asynccnt0();
    __syncthreads();

    // ---- Gram: G16 = M * M^T via 64 x v_wmma_f32_16x16x4_f32 ----
    // For D = M*M^T the A-fragment and B-fragment register layouts coincide:
    // lane l holds row (l&15), K-pair (k0 + 2*(l>>4)) in its two VGPRs.
    const int half = lane >> 4;          // 0: lanes 0-15, 1: lanes 16-31
    const int lrow = lane & 15;
    v8f acc = {};
    #pragma unroll 8
    for (int k0 = 0; k0 < 256; k0 += 4) {
        v2f f = *(const v2f*)&sM[wid][lrow][k0 + 2 * half];
        acc = __builtin_amdgcn_wmma_f32_16x16x4_f32(
            false, f, false, f, (short)0, acc, false, false);
    }
    // C/D layout: VGPR v, lanes 0-15 -> M=v ; lanes 16-31 -> M=v+8 ; N=lane&15.
    #pragma unroll
    for (int v = 0; v < 8; ++v)
        sG[wid][v + 8 * half][lrow] = acc[v];

    // Zero W16 (off-diagonal cross-patch blocks must stay 0).
    #pragma unroll
    for (int v = 0; v < 8; ++v)
        (&sW[wid][0][0])[v * 32 + lane] = 0.0f;
    __syncthreads();

    // ---- Jacobi eigensolve of each 8x8 Gram block; build W = Q f(L) Q^T ----
    if (lane < 2) {
        const int b = lane * 8;          // block offset (patch 0 / patch 1)
        float g[8][8], qm[8][8];
        #pragma unroll 1
        for (int i = 0; i < 8; ++i)
            #pragma unroll 1
            for (int j = 0; j < 8; ++j) {
                g[i][j]  = sG[wid][b + i][b + j];
                qm[i][j] = (i == j) ? 1.0f : 0.0f;
            }
        #pragma unroll 1
        for (int sweep = 0; sweep < 10; ++sweep) {
            #pragma unroll 1
            for (int p = 0; p < 7; ++p) {
                #pragma unroll 1
                for (int qi = p + 1; qi < 8; ++qi) {
                    float apq = g[p][qi];
                    if (fabsf(apq) < 1e-12f) continue;
                    float tau = (g[qi][qi] - g[p][p]) / (2.0f * apq);
                    float t   = ((tau >= 0.0f) ? 1.0f : -1.0f) /
                                (fabsf(tau) + sqrtf(1.0f + tau * tau));
                    float c   = 1.0f / sqrtf(1.0f + t * t);
                    float s   = t * c;
                    #pragma unroll 1
                    for (int k = 0; k < 8; ++k) {      // A <- A J (columns p,q)
                        float akp = g[k][p], akq = g[k][qi];
                        g[k][p]  = c * akp - s * akq;
                        g[k][qi] = s * akp + c * akq;
                    }
                    #pragma unroll 1
                    for (int k = 0; k < 8; ++k) {      // A <- J^T A (rows p,q)
                        float apk = g[p][k], aqk = g[qi][k];
                        g[p][k]  = c * apk - s * aqk;
                        g[qi][k] = s * apk + c * aqk;
                    }
                    #pragma unroll 1
                    for (int k = 0; k < 8; ++k) {      // Q <- Q J
                        float vkp = qm[k][p], vkq = qm[k][qi];
                        qm[k][p]  = c * vkp - s * vkq;
                        qm[k][qi] = s * vkp + c * vkq;
                    }
                }
            }
        }
        float fac[8];
        #pragma unroll 1
        for (int i = 0; i < 8; ++i) {
            float lam = fmaxf(g[i][i], 0.0f);
            float sig = sqrtf(lam);
            fac[i] = (sig > 1e-20f) ? fmaxf(sig - THRESH, 0.0f) / sig : 0.0f;
        }
        #pragma unroll 1
        for (int i = 0; i < 8; ++i)
            #pragma unroll 1
            for (int j = 0; j < 8; ++j) {
                float wv = 0.0f;
                #pragma unroll 1
                for (int k = 0; k < 8; ++k)
                    wv += qm[i][k] * fac[k] * qm[j][k];
                sW[wid][b + i][b + j] = wv;
            }
    }
    __syncthreads();

    // ---- Reconstruct M' = W16 * M, 16-column chunks, K=16 in 4 WMMA steps ----
    #pragma unroll 1
    for (int c0 = 0; c0 < 256; c0 += 16) {
        v8f acc2 = {};
        #pragma unroll
        for (int kk = 0; kk < 16; kk += 4) {
            v2f af = *(const v2f*)&sW[wid][lrow][kk + 2 * half];   // A = W cols
            v2f bf;                                                // B = M rows
            bf.x = sM[wid][kk + 2 * half][c0 + lrow];
            bf.y = sM[wid][kk + 2 * half + 1][c0 + lrow];
            acc2 = __builtin_amdgcn_wmma_f32_16x16x4_f32(
                false, af, false, bf, (short)0, acc2, false, false);
        }
        // Overlap-add scatter: rolled coord p writes out[(p - SHIFT) & 511].
        const int dy = c0 >> 4;
        const int dx = lrow;
        const int hp0 = wrap512(h00 + dy - SHIFT);
        const int hp1 = wrap512(h01 + dy - SHIFT);
        const int wq0 = wrap512(w00 + dx - SHIFT);
        const int wq1 = wrap512(w01 + dx - SHIFT);
        #pragma unroll
        for (int v = 0; v < 8; ++v) {
            int m  = v + 8 * half;
            int pt = m >> 3;
            int a  = m & 7;
            int n  = pt ? n1 : n0;
            int hp = pt ? hp1 : hp0;
            int wp = pt ? wq1 : wq0;
            size_t idx = ((size_t)(n * AA + a) * HH + hp) * WW + wp;
            atomicAdd(&out[idx], acc2[v]);
        }
    }
}

__global__ void llr_div_kernel(float* __restrict__ out,
                               const float* __restrict__ bw, int n) {
    int i = blockIdx.x * blockDim.x + threadIdx.x;
    if (i >= n) return;
    int w = i & 511;
    int h = (i >> 9) & 511;
    // final[q] = accum_rolled[q+SHIFT] / bw[q+SHIFT]; accum already lives at
    // final coords, so divide by the shifted weight.
    float d = bw[(size_t)wrap512(h + SHIFT) * WW + wrap512(w + SHIFT)];
    out[i] /= d;
}

extern "C" void kernel_launch(void* const* d_in, const int* in_sizes, int n_in,
                              void* d_out, int out_size, void* d_ws, size_t ws_size,
                              hipStream_t stream) {
    const float* x  = (const float*)d_in[0];   // (4, 8, 512, 512) f32
    const float* bw = (const float*)d_in[1];   // (512, 512) f32
    float* out = (float*)d_out;                // (4, 8, 512, 512) f32

    const int total = NN * AA * HH * WW;       // 8,388,608
    llr_zero_kernel<<<(total + 255) / 256, 256, 0, stream>>>(out, total);
    llr_main_kernel<<<PAIRS / WPB, WPB * 32, 0, stream>>>(x, out);
    llr_div_kernel<<<(total + 255) / 256, 256, 0, stream>>>(out, bw, total);
}